// AdaptiveCLPLLoss_2774548873720
// MI455X (gfx1250) — compile-verified
//
#include <hip/hip_runtime.h>

// Problem constants (match reference).
#define B_SIZE   2048
#define C_SIZE   50000
#define K_CAND   10
#define HEAD     2000
#define NSAMP    100
#define TAILN    (C_SIZE - HEAD)   // 48000
#define LCLIP    20.0f
#define WAVES_PER_BLOCK 8

typedef __attribute__((ext_vector_type(2))) float v2f;
typedef __attribute__((ext_vector_type(8))) float v8f;

__device__ __forceinline__ float clipf(float x) {
    return fminf(fmaxf(x, -LCLIP), LCLIP);
}
// psi(u) = log1p(exp(-u));  psi(-x) = log1p(exp(x))
__device__ __forceinline__ float psi_pos(float u) { return log1pf(expf(-u)); }
__device__ __forceinline__ float psi_neg(float x) { return log1pf(expf(x)); }

// One wave32 per row. Head sum is reduced on the matrix pipe via
// v_wmma_f32_16x16x4_f32 with an all-ones B matrix (D = A*1 + C).
__global__ __launch_bounds__(256) void clpl_row_kernel(
    const float* __restrict__ logits,
    const int*   __restrict__ candidates,
    const int*   __restrict__ sampled,
    float*       __restrict__ row_loss)
{
    const int wave = threadIdx.x >> 5;
    const int lane = threadIdx.x & 31;
    const int row  = blockIdx.x * WAVES_PER_BLOCK + wave;
    const float* rowp = logits + (size_t)row * C_SIZE;

    // ---- candidate processing (uniform across the wave) ----
    int cand[K_CAND];
#pragma unroll
    for (int k = 0; k < K_CAND; ++k) cand[k] = candidates[row * K_CAND + k];

    int   ycard    = 0;
    float candSum  = 0.f;   // sum of clipped logits over distinct valid candidates
    float headCorr = 0.f;   // psi(-x) at distinct head candidates (masked out of term2)
#pragma unroll
    for (int k = 0; k < K_CAND; ++k) {
        int  c = cand[k];
        bool v = (c >= 0);
        bool dup = false;
#pragma unroll
        for (int j = 0; j < k; ++j)
            dup = dup || ((cand[j] == c) && (cand[j] >= 0));
        if (v && !dup) {
            ++ycard;
            float x = clipf(rowp[c]);
            candSum += x;
            if (c < HEAD) headCorr += psi_neg(x);
        }
    }
    const float yc    = (float)(ycard < 1 ? 1 : ycard);
    const float term1 = psi_pos(candSum / yc);

    // ---- term2 head sum via WMMA (uniform, branch-free main loop) ----
    v8f acc  = {0.f, 0.f, 0.f, 0.f, 0.f, 0.f, 0.f, 0.f};
    v2f ones = {1.f, 1.f};

    // Main loop: 30 iterations x 64 elems = j in [0, 1920); every lane's
    // j = j0 + 2*lane satisfies j+1 <= 1919+? ... max j0=1856 -> j+1 <= 1919 < HEAD,
    // wait: last main j0 = 1856? No: j0 in {0,64,...,1856} is 30 iters covering
    // up to 1856+63 = 1919 < HEAD. No masking needed -> no exec divergence.
#pragma unroll 4
    for (int j0 = 0; j0 < 1920; j0 += 64) {
        const int j = j0 + lane * 2;
        float2 xv = *reinterpret_cast<const float2*>(rowp + j);
        v2f a;
        a.x = psi_neg(clipf(xv.x));
        a.y = psi_neg(clipf(xv.y));
        // 8 args: (neg_a, A, neg_b, B, c_mod, C, reuse_a, reuse_b)
        acc = __builtin_amdgcn_wmma_f32_16x16x4_f32(
            false, a, false, ones, (short)0, acc, false, false);
    }
    // Epilogue: 2 masked iterations covering j in [1920, 2048); reads up to
    // col 2047 are in-bounds of the 50000-wide row. Mask via float multiply
    // (lowers to cndmask+mul, keeps EXEC all-ones for the WMMA).
#pragma unroll
    for (int j0 = 1920; j0 < 2048; j0 += 64) {
        const int j = j0 + lane * 2;
        float2 xv = *reinterpret_cast<const float2*>(rowp + j);
        const float m0 = (j     < HEAD) ? 1.f : 0.f;
        const float m1 = (j + 1 < HEAD) ? 1.f : 0.f;
        v2f a;
        a.x = m0 * psi_neg(clipf(xv.x));
        a.y = m1 * psi_neg(clipf(xv.y));
        acc = __builtin_amdgcn_wmma_f32_16x16x4_f32(
            false, a, false, ones, (short)0, acc, false, false);
    }

    // D[m,n] is column-replicated (B == ones). Column N=0 lives in lane 0
    // (M=0..7) and lane 16 (M=8..15): their 8 accumulators sum to the head total.
    float lsum = acc[0] + acc[1] + acc[2] + acc[3] +
                 acc[4] + acc[5] + acc[6] + acc[7];
    float contrib = (lane == 0 || lane == 16) ? lsum : 0.f;

    // ---- term3: sampled tail (divergence OK now, WMMA is done) ----
    float t3 = 0.f;
    for (int s = lane; s < NSAMP; s += 32) {
        const int col = HEAD + sampled[s];
        bool isc = false;
#pragma unroll
        for (int k = 0; k < K_CAND; ++k)
            isc = isc || ((cand[k] >= 0) && (cand[k] == col));
        if (!isc) t3 += psi_neg(clipf(rowp[col]));
    }

    // ---- wave reduction (wave32) ----
    float v = contrib + t3 * ((float)TAILN / (float)NSAMP);
#pragma unroll
    for (int off = 16; off > 0; off >>= 1)
        v += __shfl_xor(v, off, 32);

    if (lane == 0)
        row_loss[row] = term1 - headCorr + v;
}

// Deterministic mean over the 2048 per-row losses (single block).
__global__ __launch_bounds__(256) void clpl_mean_kernel(
    const float* __restrict__ rl, float* __restrict__ out)
{
    __shared__ float sm[256];
    float s = 0.f;
    for (int i = threadIdx.x; i < B_SIZE; i += 256) s += rl[i];
    sm[threadIdx.x] = s;
    __syncthreads();
#pragma unroll
    for (int off = 128; off > 0; off >>= 1) {
        if (threadIdx.x < off) sm[threadIdx.x] += sm[threadIdx.x + off];
        __syncthreads();
    }
    if (threadIdx.x == 0) out[0] = sm[0] / (float)B_SIZE;
}

extern "C" void kernel_launch(void* const* d_in, const int* in_sizes, int n_in,
                              void* d_out, int out_size, void* d_ws, size_t ws_size,
                              hipStream_t stream) {
    (void)in_sizes; (void)n_in; (void)out_size; (void)ws_size;
    const float* logits     = (const float*)d_in[0];
    const int*   candidates = (const int*)d_in[1];   // int per harness convention
    const int*   sampled    = (const int*)d_in[2];
    float*       out        = (float*)d_out;
    float*       rowloss    = (float*)d_ws;          // 2048 floats of scratch

    clpl_row_kernel<<<B_SIZE / WAVES_PER_BLOCK, 256, 0, stream>>>(
        logits, candidates, sampled, rowloss);
    clpl_mean_kernel<<<1, 256, 0, stream>>>(rowloss, out);
}